// DecodePredictions_9423158247476
// MI455X (gfx1250) — compile-verified
//
#include <hip/hip_runtime.h>
#include <cstdint>
#include <cstddef>

#define NT 256

constexpr int   B_      = 2;
constexpr int   A_      = 196416;
constexpr int   C_      = 80;
constexpr int   K_      = 500;
constexpr int   MAXDET_ = 1000;
constexpr float CONF_   = 0.05f;
constexpr float LOGIT_CONF = -2.9444389791664403f;   // log(0.05/0.95)
constexpr float IOU_THR_   = 0.5f;

constexpr int   BINS   = 1024;
constexpr float XMIN   = -2.95f;                     // logit histogram range start
constexpr float BSCALE = BINS / 12.0f;               // covers logits up to ~9
constexpr int   NC1    = 2048;                       // per-class candidate capacity
constexpr int   NC2    = 4096;                       // per-batch merge capacity

// ---------------- CDNA5 async global->LDS helpers ----------------
#if defined(__has_builtin)
# if __has_builtin(__builtin_amdgcn_global_load_async_to_lds_b32)
#  define HAVE_ASYNC_LDS 1
# endif
# if __has_builtin(__builtin_amdgcn_s_wait_asynccnt)
#  define HAVE_WAIT_ASYNC 1
# endif
#endif

__device__ __forceinline__ void async_ld_b32(void* lds, const void* glob) {
#if defined(HAVE_ASYNC_LDS)
  // Builtin signature (per hipcc diagnostic) uses AS-qualified int pointers:
  // arg0: __device__ int* (AS1 global src), arg1: __shared__ int* (AS3 LDS dst),
  // then imm offset, imm cpol.
  typedef __attribute__((address_space(1))) int as1_int;
  typedef __attribute__((address_space(3))) int as3_int;
  __builtin_amdgcn_global_load_async_to_lds_b32(
      (as1_int*)(size_t)glob,
      (as3_int*)(size_t)(uint32_t)(size_t)lds,
      0, 0);
#else
  asm volatile("global_load_async_to_lds_b32 %0, %1, off"
               :: "v"((uint32_t)(size_t)lds), "v"(glob) : "memory");
#endif
}

template <int N>
__device__ __forceinline__ void wait_async() {
#if defined(HAVE_WAIT_ASYNC)
  __builtin_amdgcn_s_wait_asynccnt(N);
#else
  asm volatile("s_wait_asynccnt %0" :: "n"(N) : "memory");
#endif
  asm volatile("" ::: "memory");   // keep LDS reads below the wait
}

// ---------------- float <-> sortable-uint ----------------
__device__ __forceinline__ unsigned ordf(float f) {
  unsigned u = __float_as_uint(f);
  return u ^ ((u >> 31) ? 0xFFFFFFFFu : 0x80000000u);
}
__device__ __forceinline__ float unordf(unsigned u) {
  unsigned v = (u & 0x80000000u) ? (u ^ 0x80000000u) : ~u;
  return __uint_as_float(v);
}
__device__ __forceinline__ int logit_bin(float x) {
  int b = (int)((x - XMIN) * BSCALE);
  return b < 0 ? 0 : (b > BINS - 1 ? BINS - 1 : b);
}
__device__ __forceinline__ int score_bin(float s) {
  int b = (int)(s * (float)BINS);
  return b < 1 ? 1 : (b > BINS - 1 ? BINS - 1 : b);
}

// ============ K1: per-(b,c) top-K select via histogram + bitonic sort ============
__global__ __launch_bounds__(NT)
void k_topk(const float* __restrict__ cls,
            int* __restrict__ top_idx, float* __restrict__ top_val) {
  const int bc = blockIdx.x;
  const int b  = bc / C_, c = bc % C_;
  const int t  = threadIdx.x;

  __shared__ unsigned hist[BINS];
  __shared__ float stage[2][1024];
  __shared__ unsigned long long cand[NC1];
  __shared__ unsigned s_cnt;
  __shared__ int s_cb;

  for (int i = t; i < BINS; i += NT) hist[i] = 0;
  if (t == 0) s_cnt = 0;
  __syncthreads();

  const float* col = cls + (size_t)b * A_ * C_ + c;
  constexpr int CH = 1024;
  const int nch = (A_ + CH - 1) / CH;

  // --- pass 1: async double-buffered stream -> LDS -> histogram ---
  {
#pragma unroll
    for (int r = 0; r < CH / NT; ++r) {
      int a = r * NT + t; if (a >= A_) a = A_ - 1;
      async_ld_b32(&stage[0][r * NT + t], col + (size_t)a * C_);
    }
  }
  for (int ch = 0; ch < nch; ++ch) {
    if (ch + 1 < nch) {
      const int nb = (ch + 1) & 1;
#pragma unroll
      for (int r = 0; r < CH / NT; ++r) {
        int a = (ch + 1) * CH + r * NT + t; if (a >= A_) a = A_ - 1;
        async_ld_b32(&stage[nb][r * NT + t], col + (size_t)a * C_);
      }
      wait_async<4>();   // previous chunk's 4 per-wave issues have landed
    } else {
      wait_async<0>();
    }
    const int buf = ch & 1;
#pragma unroll
    for (int r = 0; r < CH / NT; ++r) {
      const int e = r * NT + t;
      const int a = ch * CH + e;
      const float x = stage[buf][e];
      if (a < A_ && x > LOGIT_CONF)
        atomicAdd(&hist[logit_bin(x)], 1u);
    }
  }
  __syncthreads();

  if (t == 0) {
    unsigned cum = 0; int cb = 0;
    for (int i = BINS - 1; i >= 0; --i) {
      cum += hist[i];
      if (cum >= (unsigned)K_) { cb = i; break; }
    }
    s_cb = cb;
  }
  __syncthreads();
  const int cb = s_cb;

  // --- pass 2: compact survivors (L2-resident reload) ---
  for (int a = t; a < A_; a += NT) {
    if (a + 4 * NT < A_) __builtin_prefetch(col + (size_t)(a + 4 * NT) * C_, 0, 1);
    const float x = col[(size_t)a * C_];
    if (x > LOGIT_CONF && logit_bin(x) >= cb) {
      unsigned p = atomicAdd(&s_cnt, 1u);
      if (p < NC1)
        cand[p] = ((unsigned long long)ordf(x) << 32) |
                  (unsigned long long)(0xFFFFFFFFu - (unsigned)a);
    }
  }
  __syncthreads();
  unsigned n = s_cnt; if (n > NC1) n = NC1;
  for (unsigned i = t; i < NC1; i += NT) if (i >= n) cand[i] = 0ull;

  // --- bitonic sort, descending ---
  for (unsigned k = 2; k <= (unsigned)NC1; k <<= 1)
    for (unsigned j = k >> 1; j > 0; j >>= 1) {
      __syncthreads();
      for (unsigned i = t; i < (unsigned)NC1; i += NT) {
        unsigned ixj = i ^ j;
        if (ixj > i) {
          unsigned long long a0 = cand[i], b0 = cand[ixj];
          if ((a0 < b0) == ((i & k) == 0)) { cand[i] = b0; cand[ixj] = a0; }
        }
      }
    }
  __syncthreads();

  for (int s = t; s < K_; s += NT) {
    unsigned long long kk = cand[s];
    float sc = 0.f; int a = 0;
    if (kk) {
      float x = unordf((unsigned)(kk >> 32));
      sc = 1.f / (1.f + __expf(-x));
      a  = (int)(0xFFFFFFFFu - (unsigned)(kk & 0xFFFFFFFFu));
    }
    top_val[(size_t)bc * K_ + s] = sc;
    top_idx[(size_t)bc * K_ + s] = a;
  }
}

// ============ K2: decode boxes + exact greedy NMS per (b,c) ============
__global__ __launch_bounds__(NT)
void k_nms(const float* __restrict__ reg, const float* __restrict__ anc,
           const int* __restrict__ top_idx, const float* __restrict__ top_val,
           float* __restrict__ cand_box, float* __restrict__ kept) {
  const int bc = blockIdx.x;
  const int b  = bc / C_;
  const int t  = threadIdx.x;

  __shared__ float bx[K_][4];
  __shared__ float ar[K_];
  __shared__ float sc[K_];
  __shared__ unsigned keepw[(K_ + 31) / 32];

  for (int i = t; i < (K_ + 31) / 32; i += NT) keepw[i] = 0u;
  __syncthreads();

  for (int s = t; s < K_; s += NT) {
    const int   a   = top_idx[(size_t)bc * K_ + s];
    const float scv = top_val[(size_t)bc * K_ + s];
    const float4 r  = ((const float4*)reg)[(size_t)b * A_ + a];
    const float4 an = ((const float4*)anc)[a];
    const float t0 = r.x * 0.1f, t1 = r.y * 0.1f, t2 = r.z * 0.2f, t3 = r.w * 0.2f;
    const float cx = t0 * an.z + an.x, cy = t1 * an.w + an.y;
    const float w  = __expf(t2) * an.z, h = __expf(t3) * an.w;
    const float x1 = cx - 0.5f * w, y1 = cy - 0.5f * h;
    const float x2 = cx + 0.5f * w, y2 = cy + 0.5f * h;
    bx[s][0] = x1; bx[s][1] = y1; bx[s][2] = x2; bx[s][3] = y2;
    ar[s] = (x2 - x1) * (y2 - y1);
    sc[s] = scv;
    ((float4*)cand_box)[(size_t)bc * K_ + s] = make_float4(x1, y1, x2, y2);
    if (scv > CONF_) atomicOr(&keepw[s >> 5], 1u << (s & 31));
  }
  __syncthreads();

  for (int i = 0; i < K_; ++i) {
    const bool on = (keepw[i >> 5] >> (i & 31)) & 1u;   // uniform across block
    if (on) {
      const float x1i = bx[i][0], y1i = bx[i][1];
      const float x2i = bx[i][2], y2i = bx[i][3], ai = ar[i];
      for (int j = i + 1 + t; j < K_; j += NT) {
        const float xx1 = fmaxf(x1i, bx[j][0]);
        const float yy1 = fmaxf(y1i, bx[j][1]);
        const float xx2 = fminf(x2i, bx[j][2]);
        const float yy2 = fminf(y2i, bx[j][3]);
        const float w = fmaxf(xx2 - xx1, 0.f), h = fmaxf(yy2 - yy1, 0.f);
        const float inter = w * h;
        const float iou = inter / fmaxf(ai + ar[j] - inter, 1e-8f);
        if (iou > IOU_THR_) atomicAnd(&keepw[j >> 5], ~(1u << (j & 31)));
      }
    }
    __syncthreads();
  }

  for (int s = t; s < K_; s += NT) {
    const bool kp = (keepw[s >> 5] >> (s & 31)) & 1u;
    kept[(size_t)bc * K_ + s] = kp ? sc[s] : 0.f;
  }
}

// ============ K3: per-batch merge, global top-MAXDET ============
__global__ __launch_bounds__(NT)
void k_merge(const float* __restrict__ kept, const float* __restrict__ cand_box,
             float* __restrict__ out_boxes, float* __restrict__ out_scores,
             float* __restrict__ out_classes, int* __restrict__ out_valid) {
  const int b = blockIdx.x;
  const int t = threadIdx.x;
  constexpr int N = C_ * K_;   // 40000

  __shared__ unsigned hist[BINS];
  __shared__ unsigned long long cand[NC2];
  __shared__ unsigned s_cnt, s_pos;
  __shared__ int s_cb;

  for (int i = t; i < BINS; i += NT) hist[i] = 0;
  if (t == 0) { s_cnt = 0; s_pos = 0; }
  __syncthreads();

  const float* ks = kept + (size_t)b * N;
  for (int n = t; n < N; n += NT) {
    const float s = ks[n];
    if (s > 0.f) {
      atomicAdd(&s_pos, 1u);
      atomicAdd(&hist[score_bin(s)], 1u);
    }
  }
  __syncthreads();

  if (t == 0) {
    unsigned cum = 0; int cb = 1;
    for (int i = BINS - 1; i >= 1; --i) {
      cum += hist[i];
      if (cum >= (unsigned)MAXDET_) { cb = i; break; }
    }
    s_cb = cb;
  }
  __syncthreads();
  const int cb = s_cb;

  for (int n = t; n < N; n += NT) {
    const float s = ks[n];
    if (s > 0.f && score_bin(s) >= cb) {
      unsigned p = atomicAdd(&s_cnt, 1u);
      if (p < NC2)
        cand[p] = ((unsigned long long)ordf(s) << 32) |
                  (unsigned long long)(0xFFFFFFFFu - (unsigned)n);
    }
  }
  __syncthreads();
  unsigned cnt = s_cnt; if (cnt > NC2) cnt = NC2;
  for (unsigned i = t; i < NC2; i += NT) if (i >= cnt) cand[i] = 0ull;

  for (unsigned k = 2; k <= (unsigned)NC2; k <<= 1)
    for (unsigned j = k >> 1; j > 0; j >>= 1) {
      __syncthreads();
      for (unsigned i = t; i < (unsigned)NC2; i += NT) {
        unsigned ixj = i ^ j;
        if (ixj > i) {
          unsigned long long a0 = cand[i], b0 = cand[ixj];
          if ((a0 < b0) == ((i & k) == 0)) { cand[i] = b0; cand[ixj] = a0; }
        }
      }
    }
  __syncthreads();

  for (int s = t; s < MAXDET_; s += NT) {
    const unsigned long long kk = cand[s];
    float fs = 0.f; int n = 0;
    if (kk) {
      fs = unordf((unsigned)(kk >> 32));
      n  = (int)(0xFFFFFFFFu - (unsigned)(kk & 0xFFFFFFFFu));
    }
    const bool ok = fs > 0.f;
    const size_t o = (size_t)b * MAXDET_ + s;
    float4 bb = make_float4(0.f, 0.f, 0.f, 0.f);
    if (ok) bb = ((const float4*)cand_box)[(size_t)b * N + n];
    ((float4*)out_boxes)[o] = bb;
    out_scores[o]  = ok ? fs : 0.f;
    out_classes[o] = ok ? (float)(n / K_) : 0.f;
  }
  if (t == 0) out_valid[b] = (int)(s_pos < (unsigned)MAXDET_ ? s_pos : (unsigned)MAXDET_);
}

// ============ host launcher ============
extern "C" void kernel_launch(void* const* d_in, const int* in_sizes, int n_in,
                              void* d_out, int out_size, void* d_ws, size_t ws_size,
                              hipStream_t stream) {
  (void)in_sizes; (void)n_in; (void)out_size; (void)ws_size;
  const float* cls = (const float*)d_in[0];   // [B,A,C] logits
  const float* reg = (const float*)d_in[1];   // [B,A,4]
  const float* anc = (const float*)d_in[2];   // [A,4] (cx,cy,w,h)

  char* ws = (char*)d_ws;
  size_t off = 0;
  auto suballoc = [&](size_t bytes) -> void* {
    void* p = ws + off;
    off = (off + bytes + 255) & ~(size_t)255;
    return p;
  };
  int*   top_idx  = (int*)  suballoc(sizeof(int)   * (size_t)B_ * C_ * K_);
  float* top_val  = (float*)suballoc(sizeof(float) * (size_t)B_ * C_ * K_);
  float* cand_box = (float*)suballoc(sizeof(float) * (size_t)B_ * C_ * K_ * 4);
  float* kept     = (float*)suballoc(sizeof(float) * (size_t)B_ * C_ * K_);

  float* out         = (float*)d_out;
  float* out_boxes   = out;                                  // [B,MAXDET,4]
  float* out_scores  = out + (size_t)B_ * MAXDET_ * 4;       // [B,MAXDET]
  float* out_classes = out_scores + (size_t)B_ * MAXDET_;    // [B,MAXDET]
  int*   out_valid   = (int*)(out_classes + (size_t)B_ * MAXDET_);  // [B]

  k_topk <<<B_ * C_, NT, 0, stream>>>(cls, top_idx, top_val);
  k_nms  <<<B_ * C_, NT, 0, stream>>>(reg, anc, top_idx, top_val, cand_box, kept);
  k_merge<<<B_,      NT, 0, stream>>>(kept, cand_box, out_boxes, out_scores,
                                      out_classes, out_valid);
}